// ConstructAdjMatrix_8031588843576
// MI455X (gfx1250) — compile-verified
//
#include <hip/hip_runtime.h>

#define N_CELL 8192
#define N_DRUG 4096

typedef __attribute__((ext_vector_type(2))) float v2f;
typedef __attribute__((ext_vector_type(4))) float v4f;
typedef __attribute__((ext_vector_type(8))) float v8f;

// ---------------------------------------------------------------------------
// Row sums via V_WMMA_F32_16X16X4_F32: D = A(16x4 adj tile) x B(4x16 ones) + C.
// Each wave (32 lanes) owns 16 rows; accumulates 4 K-columns per WMMA.
// A layout (f32 16x4): lanes 0-15 hold M=lane, K={0,1}; lanes 16-31 hold
// M=lane-16, K={2,3}. Two accumulators for ILP. Exact f32 adds.
// ---------------------------------------------------------------------------
__global__ __launch_bounds__(32) void rowsum_wmma(const float* __restrict__ adj,
                                                  float* __restrict__ rowsum) {
  const int lane = threadIdx.x;
  const int m = lane & 15;
  const int khalf = (lane >> 4) * 2;            // 0 or 2
  const size_t rowbase = (size_t)(blockIdx.x * 16 + m) * N_DRUG;

  v8f acc0 = {};
  v8f acc1 = {};
  v2f ones; ones.x = 1.0f; ones.y = 1.0f;

  for (int k = 0; k < N_DRUG; k += 8) {
    v2f a0, a1;
    a0.x = adj[rowbase + k + khalf];
    a0.y = adj[rowbase + k + khalf + 1];
    a1.x = adj[rowbase + k + 4 + khalf];
    a1.y = adj[rowbase + k + 4 + khalf + 1];
    acc0 = __builtin_amdgcn_wmma_f32_16x16x4_f32(false, a0, false, ones,
                                                 (short)0, acc0, false, false);
    acc1 = __builtin_amdgcn_wmma_f32_16x16x4_f32(false, a1, false, ones,
                                                 (short)0, acc1, false, false);
  }
  // C/D layout: VGPR v, lanes 0-15 -> M=v; lanes 16-31 -> M=v+8 (any N works).
  if (lane == 0) {
    #pragma unroll
    for (int v = 0; v < 8; ++v)
      rowsum[blockIdx.x * 16 + v] = acc0[v] + acc1[v];
  } else if (lane == 16) {
    #pragma unroll
    for (int v = 0; v < 8; ++v)
      rowsum[blockIdx.x * 16 + 8 + v] = acc0[v] + acc1[v];
  }
}

// ---------------------------------------------------------------------------
// Column sums: D = A(16x4 ones) x B(4x16 adj tile) + C -> D[m][n] = colsum[n].
// B layout (f32 4x16): lanes 0-15 hold N=lane, K={0,1}; lanes 16-31 hold
// N=lane-16, K={2,3}. Each wave owns 16 columns, marches down 8192 rows.
// ---------------------------------------------------------------------------
__global__ __launch_bounds__(32) void colsum_wmma(const float* __restrict__ adj,
                                                  float* __restrict__ colsum) {
  const int lane = threadIdx.x;
  const int n = blockIdx.x * 16 + (lane & 15);
  const int kb = (lane >> 4) * 2;               // 0 or 2

  v8f acc0 = {};
  v8f acc1 = {};
  v2f ones; ones.x = 1.0f; ones.y = 1.0f;

  for (int k = 0; k < N_CELL; k += 8) {
    v2f b0, b1;
    b0.x = adj[(size_t)(k + kb) * N_DRUG + n];
    b0.y = adj[(size_t)(k + kb + 1) * N_DRUG + n];
    b1.x = adj[(size_t)(k + 4 + kb) * N_DRUG + n];
    b1.y = adj[(size_t)(k + 4 + kb + 1) * N_DRUG + n];
    acc0 = __builtin_amdgcn_wmma_f32_16x16x4_f32(false, ones, false, b0,
                                                 (short)0, acc0, false, false);
    acc1 = __builtin_amdgcn_wmma_f32_16x16x4_f32(false, ones, false, b1,
                                                 (short)0, acc1, false, false);
  }
  // Every row of D holds colsum; lanes 0-15 read D[0][lane] from VGPR 0.
  if (lane < 16) colsum[n] = acc0[0] + acc1[0];
}

// dx = (rowsum+1)^-0.5, dy = (colsum+1)^-0.5
__global__ void norm_kernel(const float* __restrict__ rowsum,
                            const float* __restrict__ colsum,
                            float* __restrict__ dx, float* __restrict__ dy) {
  int t = blockIdx.x * blockDim.x + threadIdx.x;
  if (t < N_CELL) dx[t] = 1.0f / sqrtf(rowsum[t] + 1.0f);
  if (t < N_DRUG) dy[t] = 1.0f / sqrtf(colsum[t] + 1.0f);
}

// ---------------------------------------------------------------------------
// Fused scale + transpose: read adj once, write dx*adj*dy (row-major) and its
// transpose. 32x32 LDS tile (+1 pad) for coalesced transposed stores.
// Non-temporal stores: 604 MB of output should not displace L2.
// ---------------------------------------------------------------------------
__global__ __launch_bounds__(256) void scale_kernel(const float* __restrict__ adj,
                                                    const float* __restrict__ dx,
                                                    const float* __restrict__ dy,
                                                    float* __restrict__ out0,
                                                    float* __restrict__ out1) {
  __shared__ float tile[32][33];
  const int tx = threadIdx.x;        // 0..31
  const int ty = threadIdx.y;        // 0..7
  const int row0 = blockIdx.y * 32;
  const int col0 = blockIdx.x * 32;

  const float dyv = dy[col0 + tx];
  #pragma unroll
  for (int r = ty; r < 32; r += 8) {
    const int i = row0 + r;
    const int j = col0 + tx;
    const float s = dx[i] * adj[(size_t)i * N_DRUG + j] * dyv;
    __builtin_nontemporal_store(s, &out0[(size_t)i * N_DRUG + j]);
    tile[r][tx] = s;
  }
  __syncthreads();
  #pragma unroll
  for (int r = ty; r < 32; r += 8) {
    const int j = col0 + r;          // drug index = out1 row
    const int i = row0 + tx;         // cell index = out1 col
    __builtin_nontemporal_store(tile[tx][r], &out1[(size_t)j * N_CELL + i]);
  }
}

// ---------------------------------------------------------------------------
// Diagonal fill: one block per row, 16-byte zero-stream with the diagonal
// element patched in-register before the store. diag = 1/(sum+1) + 1.
// ---------------------------------------------------------------------------
__global__ __launch_bounds__(256) void diag_fill(float* __restrict__ out,
                                                 const float* __restrict__ sums,
                                                 int n) {
  const int i = blockIdx.x;                       // row
  const float dval = 1.0f / (sums[i] + 1.0f) + 1.0f;
  float* row = out + (size_t)i * n;
  const int diag4 = i >> 2;                       // v4f slot holding diagonal
  const int diagl = i & 3;
  const int nq = n >> 2;
  for (int q = threadIdx.x; q < nq; q += 256) {
    v4f v = {0.f, 0.f, 0.f, 0.f};
    if (q == diag4) v[diagl] = dval;
    __builtin_nontemporal_store(v, (v4f*)row + q);
  }
}

extern "C" void kernel_launch(void* const* d_in, const int* in_sizes, int n_in,
                              void* d_out, int out_size, void* d_ws, size_t ws_size,
                              hipStream_t stream) {
  const float* adj = (const float*)d_in[0];
  float* out = (float*)d_out;
  float* ws = (float*)d_ws;

  float* rowsum = ws;                 // 8192
  float* colsum = rowsum + N_CELL;    // 4096
  float* dx     = colsum + N_DRUG;    // 8192
  float* dy     = dx + N_CELL;        // 4096   (total 96 KB of ws)

  float* out0 = out;                                   // [8192,4096]
  float* out1 = out0 + (size_t)N_CELL * N_DRUG;        // [4096,8192]
  float* out2 = out1 + (size_t)N_DRUG * N_CELL;        // [8192,8192] diag
  float* out3 = out2 + (size_t)N_CELL * N_CELL;        // [4096,4096] diag

  rowsum_wmma<<<N_CELL / 16, 32, 0, stream>>>(adj, rowsum);
  colsum_wmma<<<N_DRUG / 16, 32, 0, stream>>>(adj, colsum);
  norm_kernel<<<N_CELL / 256, 256, 0, stream>>>(rowsum, colsum, dx, dy);

  dim3 sblk(32, 8), sgrd(N_DRUG / 32, N_CELL / 32);
  scale_kernel<<<sgrd, sblk, 0, stream>>>(adj, dx, dy, out0, out1);

  diag_fill<<<N_CELL, 256, 0, stream>>>(out2, rowsum, N_CELL);
  diag_fill<<<N_DRUG, 256, 0, stream>>>(out3, colsum, N_DRUG);
}